// ExLRestSelfAtten_24000277250224
// MI455X (gfx1250) — compile-verified
//
#include <hip/hip_runtime.h>
#include <math.h>

typedef __attribute__((ext_vector_type(2))) float v2f;
typedef __attribute__((ext_vector_type(4))) float v4f;
typedef __attribute__((ext_vector_type(8))) float v8f;

#define SEQ   100
#define BATCH 256
#define INDIM 300
#define HID   128
#define NROW  (BATCH*SEQ)   /* 25600 */
#define WIN   11

static __device__ __forceinline__ v8f wmma_f32_k4(v2f a, v2f b, v8f c) {
    // V_WMMA_F32_16X16X4_F32 : D = A(16x4,f32) * B(4x16,f32) + C(16x16,f32)
    return __builtin_amdgcn_wmma_f32_16x16x4_f32(false, a, false, b, (short)0, c,
                                                 false, false);
}

// ---------------------------------------------------------------------------
// Kernel 1: h = relu(x @ W1 + b1)   [25600 x 300] x [300 x 128]
// One block = one 16-row M tile; 8 waves cover the 8 N-tiles of HID=128.
// x tile staged in LDS with stride 301 (gcd(45,64)=1 -> conflict free reads).
// ---------------------------------------------------------------------------
__global__ __launch_bounds__(256) void k_h(const float* __restrict__ x,
                                           const float* __restrict__ W1,
                                           const float* __restrict__ b1,
                                           float* __restrict__ h) {
    __shared__ float xs[16 * 301];
    const int m0 = blockIdx.x * 16;
    for (int e = threadIdx.x; e < 16 * INDIM; e += 256) {
        int row = e / INDIM, col = e - row * INDIM;
        xs[row * 301 + col] = x[(m0 + row) * INDIM + col];
    }
    __syncthreads();

    const int wave = threadIdx.x >> 5;
    const int lane = threadIdx.x & 31;
    const int rr   = lane & 15;     // A: M row  / B: N col
    const int half = lane >> 4;     // K-pair select
    const int n    = wave * 16 + rr;

    v8f acc = {0.f,0.f,0.f,0.f,0.f,0.f,0.f,0.f};
    for (int kk = 0; kk < INDIM; kk += 4) {
        const int off = kk + half * 2;
        v2f a; a.x = xs[rr * 301 + off]; a.y = xs[rr * 301 + off + 1];
        v2f bf; bf.x = W1[off * HID + n]; bf.y = W1[(off + 1) * HID + n];
        acc = wmma_f32_k4(a, bf, acc);
    }
    const float bias = b1[n];
#pragma unroll
    for (int r = 0; r < 8; ++r) {
        const int m = m0 + r + half * 8;
        float v = acc[r] + bias;
        h[m * HID + n] = v > 0.f ? v : 0.f;
    }
}

// ---------------------------------------------------------------------------
// Kernel 2: fused Q = h@Wq, K = h@Wk, V = h@Wv  ([25600x128] x [128x128] x3)
// One A-fragment load feeds three WMMAs (3 accumulators per wave).
// ---------------------------------------------------------------------------
__global__ __launch_bounds__(256) void k_qkv(const float* __restrict__ h,
                                             const float* __restrict__ Wq,
                                             const float* __restrict__ Wk,
                                             const float* __restrict__ Wv,
                                             float* __restrict__ Q,
                                             float* __restrict__ K,
                                             float* __restrict__ V) {
    __shared__ float hs[16 * 132];   // stride 132 -> conflict-free column reads
    const int m0 = blockIdx.x * 16;
    for (int e = threadIdx.x; e < 16 * HID; e += 256) {
        int row = e >> 7, col = e & 127;
        hs[row * 132 + col] = h[(m0 + row) * HID + col];
    }
    __syncthreads();

    const int wave = threadIdx.x >> 5;
    const int lane = threadIdx.x & 31;
    const int rr   = lane & 15;
    const int half = lane >> 4;
    const int n    = wave * 16 + rr;

    v8f aq = {0.f,0.f,0.f,0.f,0.f,0.f,0.f,0.f};
    v8f ak = aq, av = aq;
    for (int kk = 0; kk < HID; kk += 4) {
        const int off = kk + half * 2;
        v2f a; a.x = hs[rr * 132 + off]; a.y = hs[rr * 132 + off + 1];
        v2f bq; bq.x = Wq[off * HID + n]; bq.y = Wq[(off + 1) * HID + n];
        aq = wmma_f32_k4(a, bq, aq);
        v2f bk; bk.x = Wk[off * HID + n]; bk.y = Wk[(off + 1) * HID + n];
        ak = wmma_f32_k4(a, bk, ak);
        v2f bv; bv.x = Wv[off * HID + n]; bv.y = Wv[(off + 1) * HID + n];
        av = wmma_f32_k4(a, bv, av);
    }
#pragma unroll
    for (int r = 0; r < 8; ++r) {
        const int m = m0 + r + half * 8;
        Q[m * HID + n] = aq[r];
        K[m * HID + n] = ak[r];
        V[m * HID + n] = av[r];
    }
}

// ---------------------------------------------------------------------------
// Kernel 3: per (b,s): 11x11 scores via one padded 16x16 WMMA chain, softmax,
// alpha output, pooled = sum_j (sum_i alpha_ij) * v_j, out = sigmoid(.@W2+b2)
// One wave per (b,s) position (25600 waves total).
// ---------------------------------------------------------------------------
__global__ __launch_bounds__(256) void k_attn(const float* __restrict__ Q,
                                              const float* __restrict__ Km,
                                              const float* __restrict__ V,
                                              const float* __restrict__ W2,
                                              const float* __restrict__ b2,
                                              float* __restrict__ out,
                                              float* __restrict__ alphaOut) {
    const int wave = threadIdx.x >> 5;
    const int lane = threadIdx.x & 31;
    const int rr   = lane & 15;     // A row i / B col j
    const int half = lane >> 4;
    const int idx  = blockIdx.x * 8 + wave;   // = b*SEQ + s
    if (idx >= NROW) return;
    const int s = idx % SEQ;

    // neighbor row for window index rr (used for both Q rows and K cols):
    // q[b,s,i] = Qh[b, s+5-i] ; k[b,s,j] = Kh[b, s+5-j]
    const int  r     = s + 5 - rr;
    const bool rowok = (rr < WIN) && (r >= 0) && (r < SEQ);
    const float* qrow = Q  + (long)(idx + 5 - rr) * HID;
    const float* krow = Km + (long)(idx + 5 - rr) * HID;

    v8f acc = {0.f,0.f,0.f,0.f,0.f,0.f,0.f,0.f};
    for (int kk = 0; kk < HID; kk += 4) {
        const int off = kk + half * 2;
        v2f a  = {0.f, 0.f};
        v2f bf = {0.f, 0.f};
        if (rowok) {
            a  = *(const v2f*)(qrow + off);
            bf = *(const v2f*)(krow + off);
        }
        acc = wmma_f32_k4(a, bf, acc);
    }

    const float scale = 0.08838834764831845f;  // 1/sqrt(128)
    const long  abase = (long)idx * (WIN * WIN);
    float bsum = 0.f;
#pragma unroll
    for (int rI = 0; rI < 8; ++rI) {
        const float sc = acc[rI] * scale;
        // row-wise softmax over j (16 lanes of this half-wave), mask j>=11
        float mv = (rr < WIN) ? sc : -3.0e38f;
        mv = fmaxf(mv, __shfl_xor(mv, 1, 32));
        mv = fmaxf(mv, __shfl_xor(mv, 2, 32));
        mv = fmaxf(mv, __shfl_xor(mv, 4, 32));
        mv = fmaxf(mv, __shfl_xor(mv, 8, 32));
        float p = (rr < WIN) ? expf(sc - mv) : 0.f;
        float sum = p;
        sum += __shfl_xor(sum, 1, 32);
        sum += __shfl_xor(sum, 2, 32);
        sum += __shfl_xor(sum, 4, 32);
        sum += __shfl_xor(sum, 8, 32);
        const float al = p / sum;
        const int i = rI + half * 8;
        if (i < WIN) {
            if (rr < WIN) alphaOut[abase + i * WIN + rr] = al;
            bsum += al;                       // beta_j partial (j = rr)
        }
    }
    bsum += __shfl_xor(bsum, 16, 32);         // beta_j in lanes j and j+16

    // pooled[d] = sum_j beta_j * Vh[b, s+5-j][d] ; lane owns d = lane*4..+3
    v4f pooled = {0.f, 0.f, 0.f, 0.f};
    for (int j = 0; j < WIN; ++j) {
        const float bj = __shfl(bsum, j, 32);
        const int rv = s + 5 - j;
        if (rv >= 0 && rv < SEQ) {
            v4f vv = *(const v4f*)(V + (long)(idx + 5 - j) * HID + lane * 4);
            pooled += bj * vv;
        }
    }

    // out = sigmoid(pooled @ W2 + b2), W2 is [128 x 2]
    float p0 = 0.f, p1 = 0.f;
#pragma unroll
    for (int t = 0; t < 4; ++t) {
        const int d = lane * 4 + t;
        p0 += pooled[t] * W2[d * 2 + 0];
        p1 += pooled[t] * W2[d * 2 + 1];
    }
#pragma unroll
    for (int mask = 1; mask < 32; mask <<= 1) {
        p0 += __shfl_xor(p0, mask, 32);
        p1 += __shfl_xor(p1, mask, 32);
    }
    if (lane == 0) {
        out[idx * 2 + 0] = 1.f / (1.f + expf(-(p0 + b2[0])));
        out[idx * 2 + 1] = 1.f / (1.f + expf(-(p1 + b2[1])));
    }
}

// ---------------------------------------------------------------------------
extern "C" void kernel_launch(void* const* d_in, const int* in_sizes, int n_in,
                              void* d_out, int out_size, void* d_ws, size_t ws_size,
                              hipStream_t stream) {
    const float* x  = (const float*)d_in[0];
    const float* W1 = (const float*)d_in[1];
    const float* b1 = (const float*)d_in[2];
    const float* Wq = (const float*)d_in[3];
    const float* Wk = (const float*)d_in[4];
    const float* Wv = (const float*)d_in[5];
    const float* W2 = (const float*)d_in[6];
    const float* b2 = (const float*)d_in[7];

    float* h = (float*)d_ws;                       // 25600*128
    float* Q = h + (size_t)NROW * HID;
    float* K = Q + (size_t)NROW * HID;
    float* V = K + (size_t)NROW * HID;

    float* out      = (float*)d_out;               // (B,SEQ,2) first
    float* alphaOut = out + (size_t)NROW * 2;      // then (B*SEQ,11,11)

    k_h  <<<NROW / 16, 256, 0, stream>>>(x, W1, b1, h);
    k_qkv<<<NROW / 16, 256, 0, stream>>>(h, Wq, Wk, Wv, Q, K, V);
    k_attn<<<NROW / 8, 256, 0, stream>>>(Q, K, V, W2, b2, out, alphaOut);
}